// Convolution_84172769067729
// MI455X (gfx1250) — compile-verified
//
#include <hip/hip_runtime.h>
#include <math.h>

// Problem constants (match the reference)
#define MULC    32
#define NNODES  50000
#define NEDGES  800000
#define NBASIS  10
#define NHID    100

typedef __attribute__((ext_vector_type(2))) float v2f;
typedef __attribute__((ext_vector_type(8))) float v8f;
typedef __attribute__((ext_vector_type(4))) unsigned int v4u;
typedef __attribute__((ext_vector_type(8))) int v8i;
typedef __attribute__((ext_vector_type(4))) int v4i;

// V_WMMA_F32_16X16X4_F32 : D(16x16 f32) = A(16x4 f32) * B(4x16 f32) + C
__device__ __forceinline__ v8f wmma4(v2f a, v2f b, v8f c) {
    return __builtin_amdgcn_wmma_f32_16x16x4_f32(false, a, false, b, (short)0, c,
                                                 false, false);
}

// Fragment index helpers (wave32):
//  A frag: vgpr j, lane l -> A[m = l&15][k = k0 + 2*(l>>4) + j]
//  B frag: vgpr j, lane l -> B[k = k0 + 2*(l>>4) + j][n = l&15]
//  D frag: vgpr r, lane l -> D[m = 8*(l>>4) + r][n = l&15]

// ---------------------------------------------------------------------------
// Kernel 1: per-node fctp for self-connection (s) and pre-gather linear (y).
// One wave per 16-node tile; k-loop outer so each A fragment is loaded once.
// ---------------------------------------------------------------------------
__global__ __launch_bounds__(32)
void node_fctp_kernel(const float* __restrict__ xin, const float* __restrict__ attr,
                      const float* __restrict__ Wsc0, const float* __restrict__ Wsc1,
                      const float* __restrict__ Wl10, const float* __restrict__ Wl11,
                      float* __restrict__ y, float* __restrict__ out)
{
    const int t0   = blockIdx.x * 16;
    const int lane = threadIdx.x;
    const int lo   = lane & 15;
    const int hi   = lane >> 4;
    const float inv = 0.17677669529663687f;   // 1/sqrt(32)
    const float c_s = 0.3826834323650898f;    // sin(pi/8)

    const float* rowp = xin + (size_t)(t0 + lo) * 128;

    // c = 0 : scalar channel x0 (input col k); c = 1..3 : vector d = c-1
    for (int c = 0; c < 4; ++c) {
        const float* Ws = (c == 0) ? Wsc0 : Wsc1;
        const float* Wy = (c == 0) ? Wl10 : Wl11;
        v8f acc_s[2] = {{}, {}};
        v8f acc_y[2] = {{}, {}};
        for (int k0 = 0; k0 < 32; k0 += 4) {
            const int k = k0 + 2 * hi;
            v2f a;
            if (c == 0) { a.x = rowp[k];            a.y = rowp[k + 1]; }
            else        { const int d = c - 1;
                          a.x = rowp[32 + 3*k + d]; a.y = rowp[32 + 3*(k+1) + d]; }
            for (int nt = 0; nt < 2; ++nt) {
                const int n = nt * 16 + lo;
                v2f bs = { Ws[k * 32 + n], Ws[(k + 1) * 32 + n] };
                v2f by = { Wy[k * 32 + n], Wy[(k + 1) * 32 + n] };
                acc_s[nt] = wmma4(a, bs, acc_s[nt]);
                acc_y[nt] = wmma4(a, by, acc_y[nt]);
            }
        }
        for (int nt = 0; nt < 2; ++nt) {
            for (int r = 0; r < 8; ++r) {
                const int node = t0 + 8 * hi + r;
                const int w    = nt * 16 + lo;
                const int col  = (c == 0) ? w : (32 + 3 * w + (c - 1));
                const float av = attr[node];
                y  [(size_t)node * 128 + col] = acc_y[nt][r] * inv * av;
                out[(size_t)node * 128 + col] = c_s * (acc_s[nt][r] * inv * av);
            }
        }
    }
}

// ---------------------------------------------------------------------------
// Kernel 2: fused per-edge MLP (WMMA) + tensor-product features + scatter-add.
// 4 waves/block, 16 edges/wave. W_fc2 staged block-wide via the Tensor Data
// Mover (overlapped with layer 1); h and w staged per-wave in LDS.
// ---------------------------------------------------------------------------
#define EDGE_WAVES 4

__global__ __launch_bounds__(EDGE_WAVES * 32)
void edge_kernel(const float* __restrict__ elem, const float* __restrict__ eattr,
                 const int* __restrict__ esrc, const int* __restrict__ edst,
                 const float* __restrict__ Wfc1, const float* __restrict__ Wfc2,
                 const float* __restrict__ y, float* __restrict__ agg)
{
    // per-wave: [16 edges][0:112) = h (padded), [112:240) = w  -> 61440 B
    __shared__ float smem[EDGE_WAVES][16][240];
    // block-wide copy of W_fc2 [100][128]                      -> 51200 B
    __shared__ float wlds[NHID * 128];

    const int wave = threadIdx.x >> 5;
    const int lane = threadIdx.x & 31;
    const int lo   = lane & 15;
    const int hi   = lane >> 4;
    const int e0   = (blockIdx.x * EDGE_WAVES + wave) * 16;

    const float inv1 = 0.31622776601683794f;  // 1/sqrt(10)
    const float inv2 = 0.1f;                  // 1/sqrt(100)

    // ---- kick off TDM: DMA W_fc2 (2-D tile 128x100, f32) into LDS ----------
    if (wave == 0) {
        const unsigned lds_addr = (unsigned)(size_t)(void*)wlds; // LDS offset = addr[31:0]
        const unsigned long long ga = (unsigned long long)(size_t)Wfc2;
        v4u g0;
        g0.x = 1u;                                        // count=1, user mode
        g0.y = lds_addr;                                  // D#.lds_addr
        g0.z = (unsigned)(ga & 0xFFFFFFFFu);              // global_addr[31:0]
        g0.w = (unsigned)((ga >> 32) & 0x1FFFFFFu)        // global_addr[56:32]
             | (2u << 30);                                // type = 2 (image)
        v8i g1;
        g1[0] = (2 << 16);            // data_size = 2 (4 bytes); no pad/iterate
        g1[1] = (int)(128u << 16);    // tensor_dim0 = 128 (bits 79:48 low half)
        g1[2] = (int)(100u << 16);    // tensor_dim1 = 100 (bits 111:80 low half)
        g1[3] = (int)(128u << 16);    // tile_dim0 = 128 (bits 127:112)
        g1[4] = 100;                  // tile_dim1 = 100, tile_dim2 = 0
        g1[5] = 128;                  // tensor_dim0_stride = 128
        g1[6] = 0;
        g1[7] = 0;
        v4i z4 = {};
#if defined(__clang_major__) && (__clang_major__ >= 23)
        v8i z8 = {};
        __builtin_amdgcn_tensor_load_to_lds(g0, g1, z4, z4, z8, 0);
#else
        __builtin_amdgcn_tensor_load_to_lds(g0, g1, z4, z4, 0);
#endif
    }

    // ---- layer 1: h = silu((elem @ Wfc1) * inv1), K=10 padded to 12 --------
    // Each lane's 6 A-fragment scalars loaded once, reused over 7 col tiles.
    {
        const float* arow = elem + (size_t)(e0 + lo) * NBASIS;
        const int ka = 2 * hi;                      // 0 or 2
        const float a0 = arow[ka];
        const float a1 = arow[ka + 1];
        const float a4 = arow[ka + 4];
        const float a5 = arow[ka + 5];
        const float a8 = (ka + 8 < NBASIS) ? arow[ka + 8] : 0.0f;
        const float a9 = (ka + 9 < NBASIS) ? arow[ka + 9] : 0.0f;
        for (int nt = 0; nt < 7; ++nt) {
            const int n = nt * 16 + lo;
            v8f acc = {};
            for (int k0 = 0; k0 < 12; k0 += 4) {
                const int k = k0 + ka;
                v2f a, b;
                a.x = (k0 == 0) ? a0 : (k0 == 4) ? a4 : a8;
                a.y = (k0 == 0) ? a1 : (k0 == 4) ? a5 : a9;
                b.x = (k     < NBASIS && n < NHID) ? Wfc1[k       * NHID + n] : 0.0f;
                b.y = (k + 1 < NBASIS && n < NHID) ? Wfc1[(k + 1) * NHID + n] : 0.0f;
                acc = wmma4(a, b, acc);
            }
            for (int r = 0; r < 8; ++r) {
                const float v = acc[r] * inv1;
                smem[wave][8 * hi + r][nt * 16 + lo] = v / (1.0f + __expf(-v));
            }
        }
    }

    // TDM data must be visible block-wide before layer 2 consumes it.
    if (wave == 0) __builtin_amdgcn_s_wait_tensorcnt(0);
    __syncthreads();

    // ---- layer 2: w = (h @ Wfc2) * inv2, K=100; k-outer, 8 accumulators ----
    {
        v8f wacc[8] = {{}, {}, {}, {}, {}, {}, {}, {}};
        for (int k0 = 0; k0 < NHID; k0 += 4) {
            const int k = k0 + 2 * hi;
            v2f a = { smem[wave][lo][k], smem[wave][lo][k + 1] };
            for (int nt = 0; nt < 8; ++nt) {
                const int n = nt * 16 + lo;
                v2f b = { wlds[k * 128 + n], wlds[(k + 1) * 128 + n] };
                wacc[nt] = wmma4(a, b, wacc[nt]);
            }
        }
        for (int nt = 0; nt < 8; ++nt)
            for (int r = 0; r < 8; ++r)
                smem[wave][8 * hi + r][112 + nt * 16 + lo] = wacc[nt][r] * inv2;
    }

    // ---- tensor-product features + scatter (lane u == channel u, MUL==32) --
    const float invs3 = 0.5773502691896258f;  // 1/sqrt(3)
    const int u = lane;
    for (int ei = 0; ei < 16; ++ei) {
        const int e   = e0 + ei;
        const int src = esrc[e];
        const int dst = edst[e];
        const float* yp = y + (size_t)src * 128;          // L2-resident gather
        const float y0s = yp[u];
        const float y1x = yp[32 + 3 * u + 0];
        const float y1y = yp[32 + 3 * u + 1];
        const float y1z = yp[32 + 3 * u + 2];
        const float w1 = smem[wave][ei][112 +       u];
        const float w2 = smem[wave][ei][112 +  32 + u];
        const float w3 = smem[wave][ei][112 +  64 + u];
        const float w4 = smem[wave][ei][112 +  96 + u];
        const float* ep = eattr + (size_t)e * 4;
        const float ea0 = ep[0], eax = ep[1], eay = ep[2], eaz = ep[3];

        const float mid0a = w1 * y0s * ea0;
        const float dot   = y1x * eax + y1y * eay + y1z * eaz;
        const float mid0b = w4 * dot * invs3;
        const float w2x0  = w2 * y0s;
        const float w3e   = w3 * ea0;

        float* ag = agg + (size_t)dst * 256;              // L2-resident atomics
        atomicAdd(ag +        u,         mid0a);
        atomicAdd(ag +  32 +  u,         mid0b);
        atomicAdd(ag +  64 + 3 * u + 0,  w2x0 * eax);
        atomicAdd(ag +  64 + 3 * u + 1,  w2x0 * eay);
        atomicAdd(ag +  64 + 3 * u + 2,  w2x0 * eaz);
        atomicAdd(ag + 160 + 3 * u + 0,  w3e * y1x);
        atomicAdd(ag + 160 + 3 * u + 1,  w3e * y1y);
        atomicAdd(ag + 160 + 3 * u + 2,  w3e * y1z);
    }
}

// ---------------------------------------------------------------------------
// Kernel 3: z = attr * (agg/4 @ W_l2) / sqrt(64); out += cos(pi/8) * z.
// One wave per 16-node tile; k-loop outer.
// ---------------------------------------------------------------------------
__global__ __launch_bounds__(32)
void node_out_kernel(const float* __restrict__ agg, const float* __restrict__ attr,
                     const float* __restrict__ Wl20, const float* __restrict__ Wl21,
                     float* __restrict__ out)
{
    const int t0   = blockIdx.x * 16;
    const int lane = threadIdx.x;
    const int lo   = lane & 15;
    const int hi   = lane >> 4;
    // cos(pi/8) * (1/sqrt(64)) * (1/sqrt(16))
    const float scale = 0.9238795325112867f * (1.0f / 32.0f);

    const float* rowp = agg + (size_t)(t0 + lo) * 256;

    for (int c = 0; c < 4; ++c) {
        const float* W = (c == 0) ? Wl20 : Wl21;
        v8f acc[2] = {{}, {}};
        for (int k0 = 0; k0 < 64; k0 += 4) {
            const int k = k0 + 2 * hi;
            v2f a;
            if (c == 0) { a.x = rowp[k];              a.y = rowp[k + 1]; }
            else        { const int d = c - 1;
                          a.x = rowp[64 + 3*k + d];   a.y = rowp[64 + 3*(k+1) + d]; }
            for (int nt = 0; nt < 2; ++nt) {
                const int n = nt * 16 + lo;
                v2f b = { W[k * 32 + n], W[(k + 1) * 32 + n] };
                acc[nt] = wmma4(a, b, acc[nt]);
            }
        }
        for (int nt = 0; nt < 2; ++nt) {
            for (int r = 0; r < 8; ++r) {
                const int node = t0 + 8 * hi + r;
                const int w    = nt * 16 + lo;
                const int col  = (c == 0) ? w : (32 + 3 * w + (c - 1));
                const size_t idx = (size_t)node * 128 + col;
                out[idx] = out[idx] + acc[nt][r] * scale * attr[node];
            }
        }
    }
}

// ---------------------------------------------------------------------------
extern "C" void kernel_launch(void* const* d_in, const int* in_sizes, int n_in,
                              void* d_out, int out_size, void* d_ws, size_t ws_size,
                              hipStream_t stream)
{
    const float* node_input = (const float*)d_in[0];
    const float* node_attr  = (const float*)d_in[1];
    const int*   edge_src   = (const int*)d_in[2];
    const int*   edge_dst   = (const int*)d_in[3];
    const float* edge_attr  = (const float*)d_in[4];
    const float* elem       = (const float*)d_in[5];
    const float* W_sc0      = (const float*)d_in[6];
    const float* W_sc1      = (const float*)d_in[7];
    const float* W_l10      = (const float*)d_in[8];
    const float* W_l11      = (const float*)d_in[9];
    const float* W_l20      = (const float*)d_in[10];
    const float* W_l21      = (const float*)d_in[11];
    const float* W_fc1      = (const float*)d_in[12];
    const float* W_fc2      = (const float*)d_in[13];
    float* out = (float*)d_out;

    // workspace: y [N,128] then agg [N,256]  (76.8 MB total)
    float* y   = (float*)d_ws;
    float* agg = y + (size_t)NNODES * 128;

    hipMemsetAsync(agg, 0, (size_t)NNODES * 256 * sizeof(float), stream);

    node_fctp_kernel<<<NNODES / 16, 32, 0, stream>>>(
        node_input, node_attr, W_sc0, W_sc1, W_l10, W_l11, y, out);

    edge_kernel<<<NEDGES / (16 * EDGE_WAVES), EDGE_WAVES * 32, 0, stream>>>(
        elem, edge_attr, edge_src, edge_dst, W_fc1, W_fc2, y, agg);

    node_out_kernel<<<NNODES / 16, 32, 0, stream>>>(
        agg, node_attr, W_l20, W_l21, out);
}